// GatingLayer_26061861552326
// MI455X (gfx1250) — compile-verified
//
#include <hip/hip_runtime.h>
#include <hip/hip_bf16.h>

// MoE gating layer for MI455X (gfx1250, wave32).
// logits = x[16384,3072] @ W[8,3072]^T + b ; softmax(8) ; top-2 ; softmax(2).
// Bandwidth-bound: 201 MB of f32 x @ 23.3 TB/s ~= 8.6 us floor, AI ~= 4 FLOP/B,
// so fp32 WMMA (V_WMMA_F32_16X16X4_F32) is the right matrix op (no traffic saved
// by downcasting). v2: whole W staged in LDS once (no mid-loop barriers),
// 4 token-tiles x 2 K-halves per block -> 256 blocks / 2048 free-running waves,
// dual accumulators for matrix-pipe ILP.

#define D_MODEL     3072
#define NUM_EXPERTS 8
#define TOKENS      16384      // 4 * 4096
#define TOPK        2
#define PADW        4          // dword pad per expert row: spreads LDS banks
#define LDSW        (D_MODEL + PADW)   // 3076 dwords per expert row
#define TOK_PER_BLK 64         // 4 tiles * 16 tokens; 8 waves (2 K-halves per tile)
#define K_HALF      (D_MODEL / 2)      // 1536

typedef __attribute__((ext_vector_type(2))) float v2f;
typedef __attribute__((ext_vector_type(4))) float v4f;
typedef __attribute__((ext_vector_type(8))) float v8f;

__global__ __launch_bounds__(256)
void moe_gate_wmma_kernel(const float* __restrict__ x,
                          const float* __restrict__ W,
                          const float* __restrict__ bias,
                          float* __restrict__ out)
{
    __shared__ float sW[NUM_EXPERTS * LDSW];          // all of W, expert-major, padded
    __shared__ float sLog[8][16][NUM_EXPERTS];        // [wave][token-in-tile][expert] partials

    const int tid   = threadIdx.x;
    const int wave  = tid >> 5;                       // wave32
    const int lane  = tid & 31;
    const int half  = lane >> 4;                      // K-half-wave select (ISA A/B layout)
    const int col   = lane & 15;                      // M row (A) / N column (B,C,D)
    const int tile  = wave & 3;                       // token tile 0..3
    const int kBase = (wave >> 2) * K_HALF;           // K range split across wave pairs

    const int tokBase = blockIdx.x * TOK_PER_BLK;
    const int tok     = tokBase + tile * 16 + col;    // token this lane streams (row M=col)

    // ---- One-time cooperative stage of all of W into LDS ----
    // 8 experts x 3072 floats; tid groups of 32 own one expert row (24 float4 each).
    {
        const int e  = tid >> 5;                      // expert row == wave id
        const int l5 = tid & 31;
        const float* gsrc = W + (size_t)e * D_MODEL;
        float* ldst = &sW[e * LDSW];
        #pragma unroll
        for (int r = 0; r < 24; ++r) {
            int kl = l5 * 4 + 128 * r;
            *(v4f*)(ldst + kl) = *(const v4f*)(gsrc + kl);
        }
    }
    __syncthreads();

    // ---- Main loop: free-running, no barriers ----
    // A: lane holds x[tok][k + 2*half + {0,1}] -> one coalesced b64 load per WMMA.
    // B: lane holds W[lane&7][k + 2*half + {0,1}] (expert cols 8..15 = duplicates of
    //    0..7 so EXEC stays all-ones through every WMMA; dup columns are ignored).
    const v2f* ax = (const v2f*)(x + (size_t)tok * D_MODEL + kBase + 2 * half);
    const float* wrow = &sW[(lane & 7) * LDSW + kBase + 2 * half];

    v8f acc0 = {}, acc1 = {};                         // two independent accum chains
    #pragma unroll 8
    for (int i = 0; i < K_HALF / 8; ++i) {            // 192 iters, 2 WMMAs each
        v2f a0 = ax[4 * i];
        v2f b0 = *(const v2f*)(wrow + 8 * i);
        acc0 = __builtin_amdgcn_wmma_f32_16x16x4_f32(
                   false, a0, false, b0, (short)0, acc0, false, false);
        v2f a1 = ax[4 * i + 2];
        v2f b1 = *(const v2f*)(wrow + 8 * i + 4);
        acc1 = __builtin_amdgcn_wmma_f32_16x16x4_f32(
                   false, a1, false, b1, (short)0, acc1, false, false);
    }
    v8f acc = acc0 + acc1;

    // ---- Spill partial logits: VGPR j holds M = j + 8*half (32-bit C/D layout) ----
    if (col < NUM_EXPERTS) {
        #pragma unroll
        for (int j = 0; j < 8; ++j)
            sLog[wave][j + 8 * half][col] = acc[j];
    }
    __syncthreads();

    // ---- One thread per token: sum K-halves, softmax(8), top-2, softmax(2) ----
    if (tid < TOK_PER_BLK) {
        const int gtok = tokBase + tid;
        const int t    = tid >> 4;                    // tile
        const int tl   = tid & 15;                    // token in tile

        float l[NUM_EXPERTS], p[NUM_EXPERTS];
        float m = -1e30f;
        #pragma unroll
        for (int e = 0; e < NUM_EXPERTS; ++e) {
            l[e] = sLog[t][tl][e] + sLog[t + 4][tl][e] + bias[e];
            m = fmaxf(m, l[e]);
        }
        float s = 0.0f;
        #pragma unroll
        for (int e = 0; e < NUM_EXPERTS; ++e) { p[e] = __expf(l[e] - m); s += p[e]; }
        const float inv = 1.0f / s;

        // strict '>' reproduces lax.top_k lowest-index tie-breaking
        int i1 = 0; float v1 = p[0];
        #pragma unroll
        for (int e = 1; e < NUM_EXPERTS; ++e) if (p[e] > v1) { v1 = p[e]; i1 = e; }
        int i2 = (i1 == 0) ? 1 : 0; float v2 = p[i2];
        #pragma unroll
        for (int e = 0; e < NUM_EXPERTS; ++e) if (e != i1 && p[e] > v2) { v2 = p[e]; i2 = e; }

        // Reference renormalizes the selected *probabilities* with a second softmax
        const float g1 = v1 * inv, g2 = v2 * inv;
        const float e2 = __expf(g2 - g1);
        const float w1 = 1.0f / (1.0f + e2);
        const float w2 = e2 * w1;

        out[2 * gtok + 0] = w1;
        out[2 * gtok + 1] = w2;
        float* oidx = out + (size_t)TOKENS * TOPK;    // indices, stored as float
        oidx[2 * gtok + 0] = (float)i1;
        oidx[2 * gtok + 1] = (float)i2;
    }
}

extern "C" void kernel_launch(void* const* d_in, const int* in_sizes, int n_in,
                              void* d_out, int out_size, void* d_ws, size_t ws_size,
                              hipStream_t stream) {
    (void)in_sizes; (void)n_in; (void)out_size; (void)d_ws; (void)ws_size;
    const float* x = (const float*)d_in[0];
    const float* W = (const float*)d_in[1];
    const float* b = (const float*)d_in[2];
    float* out = (float*)d_out;
    moe_gate_wmma_kernel<<<TOKENS / TOK_PER_BLK, 256, 0, stream>>>(x, W, b, out);
}